// MC2RowParallelLinearImpl_44856638439903
// MI455X (gfx1250) — compile-verified
//
#include <hip/hip_runtime.h>
#include <stdint.h>

typedef float v2f __attribute__((ext_vector_type(2)));
typedef float v8f __attribute__((ext_vector_type(8)));

// Fixed problem shape from the harness
constexpr int S  = 4096;
constexpr int Bb = 2;
constexpr int K  = 8192;
constexpr int N  = 2048;
constexpr int M  = S * Bb;          // 8192 flattened rows

constexpr int BM  = 128;            // block tile M
constexpr int BN  = 128;            // block tile N
constexpr int BK  = 16;             // K tile
constexpr int LDT = 20;             // padded LDS row stride (floats): mult of 4, conflict-free
constexpr int NKT = K / BK;         // 512 K tiles

// ---- CDNA5 async global->LDS (no VGPR staging), tracked by ASYNCcnt ----
__device__ __forceinline__ void async_b128(uint32_t lds_addr, uint64_t gaddr) {
    // VDST = LDS byte address (VGPR), VADDR = 64-bit global address (VGPR pair), SADDR = off
    asm volatile("global_load_async_to_lds_b128 %0, %1, off"
                 :: "v"(lds_addr), "v"(gaddr) : "memory");
}
#define WAIT_ASYNC_4() asm volatile("s_wait_asynccnt 0x4" ::: "memory")
#define WAIT_ASYNC_0() asm volatile("s_wait_asynccnt 0x0" ::: "memory")

__global__ __launch_bounds__(256, 1)
void rowparallel_gemm_wmma_f32(const float* __restrict__ X,
                               const float* __restrict__ W,
                               const float* __restrict__ bias,
                               float* __restrict__ Y)
{
    // Double-buffered tiles, both stored [row][k] (row = m for A, n for B)
    __shared__ float As[2][BM * LDT];
    __shared__ float Bs[2][BN * LDT];

    const int tid    = threadIdx.x;
    const int lane   = tid & 31;
    const int wave   = tid >> 5;
    const int laneLo = lane & 15;
    const int laneHi = lane >> 4;       // 0: k,k+1  1: k+2,k+3 (ISA f32 A/B frag layout)

    const int blockM = blockIdx.x * BM;
    const int blockN = blockIdx.y * BN;

    const int waveM = (wave & 3) * 32;  // 4 waves along M -> 2 m-tiles of 16
    const int waveN = (wave >> 2) * 64; // 2 waves along N -> 4 n-tiles of 16

    // Tile loader: 128 rows x 4 float4/row = 512 b128 transfers; each thread does 2 per matrix
    const int ldRow = tid >> 2;         // 0..63
    const int ldCol = (tid & 3) * 4;    // 0,4,8,12 floats

    const uint64_t AgBase = (uint64_t)(uintptr_t)(X + (size_t)(blockM + ldRow) * K + ldCol);
    const uint64_t BgBase = (uint64_t)(uintptr_t)(W + (size_t)(blockN + ldRow) * K + ldCol);
    // Low 32 bits of the flat pointer == LDS byte offset (aperture mapping)
    const uint32_t AsBase = (uint32_t)(uintptr_t)&As[0][ldRow * LDT + ldCol];
    const uint32_t BsBase = (uint32_t)(uintptr_t)&Bs[0][ldRow * LDT + ldCol];

    constexpr uint32_t BUF_STRIDE = (uint32_t)(BM * LDT * 4);     // bytes between buffers
    constexpr uint32_t ROW64_LDS  = (uint32_t)(64 * LDT * 4);     // 64 rows in LDS bytes
    constexpr uint64_t ROW64_GLB  = (uint64_t)64 * K * 4;         // 64 rows in global bytes

    auto issueTile = [&](int kt, int buf) {
        const uint64_t koff = (uint64_t)kt * (BK * 4);
        const uint32_t lbo  = buf ? BUF_STRIDE : 0u;
#pragma unroll
        for (int i = 0; i < 2; ++i) {
            async_b128(AsBase + lbo + (uint32_t)i * ROW64_LDS, AgBase + koff + (uint64_t)i * ROW64_GLB);
            async_b128(BsBase + lbo + (uint32_t)i * ROW64_LDS, BgBase + koff + (uint64_t)i * ROW64_GLB);
        }
    };

    v8f acc[2][4];
#pragma unroll
    for (int mt = 0; mt < 2; ++mt)
#pragma unroll
        for (int nt = 0; nt < 4; ++nt)
            acc[mt][nt] = v8f{0.f, 0.f, 0.f, 0.f, 0.f, 0.f, 0.f, 0.f};

    auto compute = [&](int buf) {
#pragma unroll
        for (int kk = 0; kk < BK; kk += 4) {
            const int kl = kk + laneHi * 2;   // lanes 16..31 take k+2,k+3
            v2f a[2], b[4];
#pragma unroll
            for (int mt = 0; mt < 2; ++mt)
                a[mt] = *(const v2f*)&As[buf][(waveM + mt * 16 + laneLo) * LDT + kl];
#pragma unroll
            for (int nt = 0; nt < 4; ++nt)
                b[nt] = *(const v2f*)&Bs[buf][(waveN + nt * 16 + laneLo) * LDT + kl];
#pragma unroll
            for (int mt = 0; mt < 2; ++mt)
#pragma unroll
                for (int nt = 0; nt < 4; ++nt)
                    acc[mt][nt] = __builtin_amdgcn_wmma_f32_16x16x4_f32(
                        false, a[mt], false, b[nt], (short)0, acc[mt][nt],
                        false, false);
        }
    };

    // Prologue: two tiles in flight (8 outstanding async ops per wave)
    issueTile(0, 0);
    issueTile(1, 1);

    for (int kt = 0; kt < NKT - 1; ++kt) {
        const int buf = kt & 1;
        WAIT_ASYNC_4();                 // oldest 4 async ops (tile kt) have landed
        __syncthreads();                // all waves' portions visible
        compute(buf);
        __syncthreads();                // everyone done reading buf
        if (kt + 2 < NKT) issueTile(kt + 2, buf);
    }
    // Peeled last tile: only its own 4 ops outstanding -> wait for 0
    WAIT_ASYNC_0();
    __syncthreads();
    compute((NKT - 1) & 1);

    // Epilogue: bias add + store (C/D layout: VGPR v -> row v (+8 for hi lanes), col = laneLo)
#pragma unroll
    for (int nt = 0; nt < 4; ++nt) {
        const int   col = blockN + waveN + nt * 16 + laneLo;
        const float bv  = bias[col];
#pragma unroll
        for (int mt = 0; mt < 2; ++mt) {
            const int rowBase = blockM + waveM + mt * 16 + laneHi * 8;
            v8f c = acc[mt][nt];
#pragma unroll
            for (int i = 0; i < 8; ++i)
                Y[(size_t)(rowBase + i) * N + col] = c[i] + bv;
        }
    }
}

extern "C" void kernel_launch(void* const* d_in, const int* in_sizes, int n_in,
                              void* d_out, int out_size, void* d_ws, size_t ws_size,
                              hipStream_t stream) {
    (void)in_sizes; (void)n_in; (void)out_size; (void)d_ws; (void)ws_size;
    const float* X  = (const float*)d_in[0];   // (S,B,K) fp32
    const float* W  = (const float*)d_in[1];   // (N,K)   fp32
    const float* bv = (const float*)d_in[2];   // (N,)    fp32
    float*       Y  = (float*)d_out;           // (S,B,N) fp32

    dim3 grid(M / BM, N / BN);                 // 64 x 16 blocks
    rowparallel_gemm_wmma_f32<<<grid, dim3(256), 0, stream>>>(X, W, bv, Y);
}